// EMAStats_87729001988667
// MI455X (gfx1250) — compile-verified
//
#include <hip/hip_runtime.h>
#include <hip/hip_bf16.h>
#include <stdint.h>

// EMAStats for MI455X (gfx1250): gather-own-logit + K-bin segment reduction.
// Pass1: async-gather own logits to LDS, LDS-privatized (n, sum) histogram, flush.
// Pass2: re-gather (L2 hit), centered dev^2 histogram vs batch mean, flush.
// Pass3: K-wide EMA finalize.
//
// CDNA5-specific paths: GLOBAL_LOAD_ASYNC_TO_LDS_B32 (+ S_WAIT_ASYNCCNT),
// DS_ADD_F32 LDS atomics, agent-scope GLOBAL_ATOMIC_ADD_F32, wave32 sizing.

#define EMA_DECAY 0.1f
#define EPS_VAR   1e-12f
#define KMAX      2048
#define BLOCK     256
#define ROWS      8     // rows gathered per thread per block

// Builtin signature (from hipcc diagnostic): first param is AS(1) int*, so:
//   void __builtin_amdgcn_global_load_async_to_lds_b32(global int*, local int*, imm int, imm int)
typedef __attribute__((address_space(1))) int gas_int;
typedef __attribute__((address_space(3))) int lds_int;

#if defined(__gfx1250__) && \
    __has_builtin(__builtin_amdgcn_global_load_async_to_lds_b32) && \
    __has_builtin(__builtin_amdgcn_s_wait_asynccnt)
#define USE_ASYNC_GATHER 1
#else
#define USE_ASYNC_GATHER 0
#endif

__device__ __forceinline__ void async_gather_f32(const float* gptr, float* lds_slot) {
#if USE_ASYNC_GATHER
    // dest LDS address is the low 32 bits of the generic pointer (as3 is 32-bit)
    __builtin_amdgcn_global_load_async_to_lds_b32(
        (gas_int*)(uintptr_t)gptr,
        (lds_int*)(uint32_t)(uintptr_t)lds_slot,
        /*offset=*/0, /*cpol=*/0);
#else
    *lds_slot = *gptr;
#endif
}

__device__ __forceinline__ void async_gather_wait() {
#if USE_ASYNC_GATHER
    __builtin_amdgcn_s_wait_asynccnt(0);
#endif
    __asm__ volatile("" ::: "memory");  // compiler barrier: LDS was written behind its back
}

__device__ __forceinline__ void atomic_add_agent_f32(float* p, float v) {
    __hip_atomic_fetch_add(p, v, __ATOMIC_RELAXED, __HIP_MEMORY_SCOPE_AGENT);
}

__global__ void ema_zero_kernel(float* ws, int n) {
    int i = blockIdx.x * blockDim.x + threadIdx.x;
    if (i < n) ws[i] = 0.0f;
}

__global__ __launch_bounds__(BLOCK)
void ema_pass1_kernel(const float* __restrict__ logits, const int* __restrict__ target,
                      float* __restrict__ ws_n, float* __restrict__ ws_s,
                      int B, int K) {
    __shared__ float sh_n[KMAX];
    __shared__ float sh_s[KMAX];
    __shared__ float sh_g[BLOCK * ROWS];

    const int tid  = threadIdx.x;
    const int base = blockIdx.x * (BLOCK * ROWS);
    const bool lds_hist = (K <= KMAX);

    if (lds_hist) {
        for (int k = tid; k < K; k += BLOCK) { sh_n[k] = 0.0f; sh_s[k] = 0.0f; }
        __syncthreads();
    }

    // Issue all gathers (independent, ASYNCcnt-tracked), then drain once.
    int tgt[ROWS];
#pragma unroll
    for (int r = 0; r < ROWS; ++r) {
        int i = base + r * BLOCK + tid;
        if (i < B) {
            int t = target[i];
            tgt[r] = t;
            async_gather_f32(logits + (size_t)i * (size_t)K + (size_t)t,
                             &sh_g[r * BLOCK + tid]);
        } else {
            tgt[r] = -1;
        }
    }
    async_gather_wait();

#pragma unroll
    for (int r = 0; r < ROWS; ++r) {
        int t = tgt[r];
        if (t >= 0) {
            float own = sh_g[r * BLOCK + tid];
            if (lds_hist) {
                atomicAdd(&sh_n[t], 1.0f);   // ds_add_f32
                atomicAdd(&sh_s[t], own);
            } else {
                atomic_add_agent_f32(&ws_n[t], 1.0f);
                atomic_add_agent_f32(&ws_s[t], own);
            }
        }
    }

    if (lds_hist) {
        __syncthreads();
        for (int k = tid; k < K; k += BLOCK) {
            float ln = sh_n[k];
            if (ln != 0.0f) {
                atomic_add_agent_f32(&ws_n[k], ln);
                atomic_add_agent_f32(&ws_s[k], sh_s[k]);
            }
        }
    }
}

__global__ __launch_bounds__(BLOCK)
void ema_pass2_kernel(const float* __restrict__ logits, const int* __restrict__ target,
                      const float* __restrict__ ws_n, const float* __restrict__ ws_s,
                      float* __restrict__ ws_ssd, int B, int K) {
    __shared__ float sh_bm[KMAX];
    __shared__ float sh_ssd[KMAX];
    __shared__ float sh_g[BLOCK * ROWS];

    const int tid  = threadIdx.x;
    const int base = blockIdx.x * (BLOCK * ROWS);
    const bool lds_hist = (K <= KMAX);

    if (lds_hist) {
        for (int k = tid; k < K; k += BLOCK) {
            float nk = ws_n[k];
            sh_bm[k]  = ws_s[k] / fmaxf(nk, 1.0f);
            sh_ssd[k] = 0.0f;
        }
        __syncthreads();
    }

    int tgt[ROWS];
#pragma unroll
    for (int r = 0; r < ROWS; ++r) {
        int i = base + r * BLOCK + tid;
        if (i < B) {
            int t = target[i];
            tgt[r] = t;
            async_gather_f32(logits + (size_t)i * (size_t)K + (size_t)t,
                             &sh_g[r * BLOCK + tid]);
        } else {
            tgt[r] = -1;
        }
    }
    async_gather_wait();

#pragma unroll
    for (int r = 0; r < ROWS; ++r) {
        int t = tgt[r];
        if (t >= 0) {
            float own = sh_g[r * BLOCK + tid];
            float bm  = lds_hist ? sh_bm[t] : (ws_s[t] / fmaxf(ws_n[t], 1.0f));
            float dev = own - bm;
            if (lds_hist) {
                atomicAdd(&sh_ssd[t], dev * dev);  // ds_add_f32
            } else {
                atomic_add_agent_f32(&ws_ssd[t], dev * dev);
            }
        }
    }

    if (lds_hist) {
        __syncthreads();
        for (int k = tid; k < K; k += BLOCK) {
            float v = sh_ssd[k];
            if (v != 0.0f) atomic_add_agent_f32(&ws_ssd[k], v);
        }
    }
}

__global__ void ema_finalize_kernel(const float* __restrict__ mean,
                                    const float* __restrict__ var,
                                    const int* __restrict__ count,
                                    const float* __restrict__ ws_n,
                                    const float* __restrict__ ws_s,
                                    const float* __restrict__ ws_ssd,
                                    float* __restrict__ out, int K) {
    int k = blockIdx.x * blockDim.x + threadIdx.x;
    if (k >= K) return;
    float n   = ws_n[k];
    float ns  = fmaxf(n, 1.0f);
    float bm  = ws_s[k] / ns;
    float bv  = ws_ssd[k] / ns;
    float m0  = mean[k];
    float v0  = var[k];
    int   c   = count[k];
    bool first = (c == 0);
    float cm = first ? bm : ((1.0f - EMA_DECAY) * m0 + EMA_DECAY * bm);
    float cv = fmaxf(first ? bv : ((1.0f - EMA_DECAY) * v0 + EMA_DECAY * bv), EPS_VAR);
    bool has = (n > 0.0f);
    out[k]         = has ? cm : m0;   // new_mean
    out[K + k]     = has ? cv : v0;   // new_var
    out[2 * K + k] = (float)c + n;    // new_count (exact in f32: <= 131172)
}

extern "C" void kernel_launch(void* const* d_in, const int* in_sizes, int n_in,
                              void* d_out, int out_size, void* d_ws, size_t ws_size,
                              hipStream_t stream) {
    const float* logits = (const float*)d_in[0];
    const int*   target = (const int*)d_in[1];
    const float* mean   = (const float*)d_in[2];
    const float* var    = (const float*)d_in[3];
    const int*   count  = (const int*)d_in[4];

    const int B = in_sizes[1];   // 131072
    const int K = in_sizes[2];   // 2048

    float* ws_n   = (float*)d_ws;       // [K]
    float* ws_s   = ws_n + K;           // [K]
    float* ws_ssd = ws_s + K;           // [K]
    float* out    = (float*)d_out;      // [3K] = mean | var | count

    ema_zero_kernel<<<(3 * K + 255) / 256, 256, 0, stream>>>(ws_n, 3 * K);

    const int rows_per_block = BLOCK * ROWS;                 // 2048
    const int nblocks = (B + rows_per_block - 1) / rows_per_block;  // 64

    ema_pass1_kernel<<<nblocks, BLOCK, 0, stream>>>(logits, target, ws_n, ws_s, B, K);
    ema_pass2_kernel<<<nblocks, BLOCK, 0, stream>>>(logits, target, ws_n, ws_s, ws_ssd, B, K);
    ema_finalize_kernel<<<(K + 255) / 256, 256, 0, stream>>>(mean, var, count,
                                                             ws_n, ws_s, ws_ssd, out, K);
}